// ScaleInvariantSphericalHarmonicsDomainActivation_16243566313947
// MI455X (gfx1250) — compile-verified
//
#include <hip/hip_runtime.h>
#include <math.h>

// ---------------------------------------------------------------------------
// ScaleInvariantSphericalHarmonicsDomainActivation — MI455X (gfx1250)
//
// Memory-bound (536 MB @ 23.3 TB/s ~= 23 us). The 16x16 complex matvecs are
// batched 16-points-at-a-time onto V_WMMA_F32_16X16X4_F32 (4 chained K=4
// steps = K=16) so the matrix pipe absorbs them and the VALU stays free for
// the streaming complex-gating / quarter-power-normalization phase.
// ---------------------------------------------------------------------------

typedef __attribute__((ext_vector_type(2))) float v2f;
typedef __attribute__((ext_vector_type(8))) float v8f;

#define CH   256
#define TT   128
#define FF   256
#define TAU  16
#define CHS  (TT * FF * 2)   // channel stride in floats (65536)
#define LDSS 34              // padded per-point LDS stride (conflict-free)

__device__ __forceinline__ v8f wmma_f32(v2f a, v2f b, v8f c) {
  // D = A(16x4) * B(4x16) + C(16x16), fp32, wave32
  return __builtin_amdgcn_wmma_f32_16x16x4_f32(
      /*neg_a=*/false, a, /*neg_b=*/false, b,
      /*c_mod=*/(short)0, c, /*reuse_a=*/false, /*reuse_b=*/false);
}

__global__ __launch_bounds__(256)
void sisha_act_kernel(const float* __restrict__ x,
                      const float* __restrict__ Wre,
                      const float* __restrict__ bre,
                      const float* __restrict__ Wim,
                      const float* __restrict__ bim,
                      float* __restrict__ out)
{
  __shared__ float ldsA[256 * LDSS];   // a_re/a_im interleaved per point

  const int tid  = threadIdx.x;        // == f (point) for elementwise phase
  const int lane = tid & 31;
  const int wave = tid >> 5;
  const int bt   = blockIdx.x;         // b*T + t
  const int b    = bt / TT;
  const int t    = bt - b * TT;
  const size_t base = ((size_t)b * CH * TT + (size_t)t) * (FF * 2); // (b,0,t,0,0)

  // ---- part 0: channels 0..15 pass through unchanged ----
  {
    const size_t p = base + (size_t)tid * 2;
#pragma unroll
    for (int c = 0; c < TAU; ++c) {
      v2f v = *(const v2f*)(x + p + (size_t)c * CHS);
      *(v2f*)(out + p + (size_t)c * CHS) = v;
    }
  }

  // ---- A fragments (x0 re/im) for this wave's two 16-point groups ----
  // f32 16x16x4 A layout: lanes 0-15 -> M=lane, K={k0,k0+1};
  //                       lanes 16-31 -> M=lane-16, K={k0+2,k0+3}
  const int n    = lane & 15;
  const int koff = (lane >= 16) ? 2 : 0;

  v2f are[2][4], aim[2][4], aimn[2][4];
#pragma unroll
  for (int g = 0; g < 2; ++g) {
    const int f = wave * 32 + g * 16 + n;
    const size_t p = base + (size_t)f * 2;
#pragma unroll
    for (int j = 0; j < 4; ++j) {
      const int k0 = 4 * j + koff;
      v2f r, i;
      r.x = x[p + (size_t)k0 * CHS];
      r.y = x[p + (size_t)(k0 + 1) * CHS];
      i.x = x[p + (size_t)k0 * CHS + 1];
      i.y = x[p + (size_t)(k0 + 1) * CHS + 1];
      are[g][j]  = r;
      aim[g][j]  = i;
      v2f in_;
      in_.x = -i.x; in_.y = -i.y;
      aimn[g][j] = in_;                // f32 WMMA has no A/B negate: pre-negate
    }
  }

  int begin = TAU;
#pragma unroll
  for (int l = 1; l < 4; ++l) {
    const int m = 2 * l + 1;

    // ---- B fragments: B[k][n] = W[n][k] (x @ W.T) ----
    const float* Wr = Wre + l * 256;
    const float* Wi = Wim + l * 256;
    v2f bR[4], bI[4];
#pragma unroll
    for (int j = 0; j < 4; ++j) {
      const int k0 = 4 * j + koff;
      v2f r, i;
      r.x = Wr[n * 16 + k0]; r.y = Wr[n * 16 + k0 + 1];
      i.x = Wi[n * 16 + k0]; i.y = Wi[n * 16 + k0 + 1];
      bR[j] = r; bI[j] = i;
    }
    // bias folded into accumulator init (C layout: N = lane&15, any M)
    const float biasRe = bre[l * 16 + n] - bim[l * 16 + n];
    const float biasIm = bre[l * 16 + n] + bim[l * 16 + n];

#pragma unroll
    for (int g = 0; g < 2; ++g) {
      v8f accRe, accIm;
#pragma unroll
      for (int j = 0; j < 8; ++j) { accRe[j] = biasRe; accIm[j] = biasIm; }

      // a_re = re@Wre.T - im@Wim.T + (bre-bim)
#pragma unroll
      for (int j = 0; j < 4; ++j) accRe = wmma_f32(are[g][j],  bR[j], accRe);
#pragma unroll
      for (int j = 0; j < 4; ++j) accRe = wmma_f32(aimn[g][j], bI[j], accRe);
      // a_im = re@Wim.T + im@Wre.T + (bre+bim)
#pragma unroll
      for (int j = 0; j < 4; ++j) accIm = wmma_f32(are[g][j],  bI[j], accIm);
#pragma unroll
      for (int j = 0; j < 4; ++j) accIm = wmma_f32(aim[g][j],  bR[j], accIm);

      // D layout: lanes 0-15 -> M=j, lanes 16-31 -> M=8+j; N = lane&15
      const int mbase = wave * 32 + g * 16 + ((lane >= 16) ? 8 : 0);
#pragma unroll
      for (int j = 0; j < 8; ++j) {
        v2f a2; a2.x = accRe[j]; a2.y = accIm[j];
        *(v2f*)&ldsA[(mbase + j) * LDSS + 2 * n] = a2;
      }
    }
    __syncthreads();

    // ---- elementwise gating + quarter-power normalization; lane == point ----
    {
      const size_t p = base + (size_t)tid * 2;
#pragma unroll
      for (int r = 0; r < TAU; ++r) {
        const float a_re = ldsA[tid * LDSS + 2 * r];
        const float a_im = ldsA[tid * LDSS + 2 * r + 1];
        float yre[7], yim[7];
        float sumsq = 0.f;
#pragma unroll
        for (int c = 0; c < m; ++c) {
          const int ch = begin + r * m + c;
          v2f xv = *(const v2f*)(x + p + (size_t)ch * CHS);
          const float vr = xv.x * a_re - xv.y * a_im;
          const float vi = xv.y * a_re + xv.x * a_im;
          yre[c] = vr; yim[c] = vi;
          sumsq += vr * vr + vi * vi;
        }
        const float nrm   = sqrtf(sumsq);
        const float scale = 1.0f / (1e-5f + sqrtf(nrm));
#pragma unroll
        for (int c = 0; c < m; ++c) {
          const int ch = begin + r * m + c;
          v2f yo; yo.x = yre[c] * scale; yo.y = yim[c] * scale;
          *(v2f*)(out + p + (size_t)ch * CHS) = yo;
        }
      }
    }
    __syncthreads();
    begin += TAU * m;
  }
}

extern "C" void kernel_launch(void* const* d_in, const int* in_sizes, int n_in,
                              void* d_out, int out_size, void* d_ws, size_t ws_size,
                              hipStream_t stream) {
  const float* x   = (const float*)d_in[0];
  const float* Wre = (const float*)d_in[1];
  const float* bre = (const float*)d_in[2];
  const float* Wim = (const float*)d_in[3];
  const float* bim = (const float*)d_in[4];
  float* out = (float*)d_out;

  const int B = 4;
  dim3 grid(B * TT);   // one block per (b, t); 256 threads cover F
  dim3 block(256);
  sisha_act_kernel<<<grid, block, 0, stream>>>(x, Wre, bre, Wim, bim, out);
}